// CustomModel_20237885899234
// MI455X (gfx1250) — compile-verified
//
#include <hip/hip_runtime.h>
#include <math.h>

typedef __attribute__((ext_vector_type(2))) float v2f;
typedef __attribute__((ext_vector_type(8))) float v8f;

#define THREADS 256
#define WAVES_PER_BLOCK (THREADS / 32)
#define MT 5   // M-tiles per wave: N = 50000 = 625 * (16*5), exact

__device__ __forceinline__ float sigm(float v) { return 1.0f / (1.0f + __expf(-v)); }

// ---------------------------------------------------------------------------
// Weight transposes (weights are tiny; done once per layer into workspace so
// the WMMA B-fragments become contiguous b64 loads).
//   Wt[g'][c][k] = W[g][k][c]  for gates g in {0,2,3}
// ---------------------------------------------------------------------------
__global__ void transpose_gates_kernel(const float* __restrict__ W, float* __restrict__ Wt,
                                       int fi, int fo)
{
    int i = blockIdx.x * blockDim.x + threadIdx.x;
    int total = 3 * fi * fo;
    if (i >= total) return;
    int gp  = i / (fi * fo);
    int rem = i - gp * fi * fo;
    int c   = rem / fi;
    int k   = rem - c * fi;
    const int gmap[3] = {0, 2, 3};
    Wt[i] = W[((size_t)gmap[gp] * fi + k) * fo + c];
}

// Wt[c][r] = W[r][c]
__global__ void transpose_kernel(const float* __restrict__ W, float* __restrict__ Wt,
                                 int rows, int cols)
{
    int i = blockIdx.x * blockDim.x + threadIdx.x;
    if (i >= rows * cols) return;
    int c = i / rows;
    int r = i - c * rows;
    Wt[i] = W[(size_t)r * cols + c];
}

// ---------------------------------------------------------------------------
// Gate GEMMs + LSTM activation fusion (templated: all strides are immediates).
// h1 = sigmoid(xW3+c3) * tanh( sigmoid(xW0+c0) * tanh(xW2+c2) )
// One wave owns an 80x16 output strip (5 M-tiles). Per k-step: explicit load
// phase (3 B + 5 A b64 fragments, clause-able with immediate offsets), then
// 15 v_wmma_f32_16x16x4_f32 with staggered loadcnt waits.
// ---------------------------------------------------------------------------
template <int FI, int FO>
__global__ __launch_bounds__(THREADS)
void gates_wmma_kernel(const float* __restrict__ x,     // n x FI
                       const float* __restrict__ Wt,    // 3 x FO x FI (gates 0,2,3 transposed)
                       const float* __restrict__ b,     // 4 x FO
                       const float* __restrict__ chebb, // 4 x FO
                       float* __restrict__ h1,          // n x FO
                       int n)
{
    const int lane   = threadIdx.x & 31;
    const int wave   = threadIdx.x >> 5;
    constexpr int NT = FO >> 4;
    const int gw     = blockIdx.x * WAVES_PER_BLOCK + wave;
    const int mtg    = gw / NT;
    const int nt     = gw - mtg * NT;
    const int m0     = mtg * (16 * MT);
    if (m0 >= n) return;                            // wave-uniform guard

    const int half = lane >> 4;                     // lanes 16-31 hold K{2,3} / rows 8-15
    const int r16  = lane & 15;
    const int bcol = nt * 16 + r16;

    const float* __restrict__ Wt0 = Wt + 0 * FO * FI + bcol * FI;
    const float* __restrict__ Wt2 = Wt + 1 * FO * FI + bcol * FI;
    const float* __restrict__ Wt3 = Wt + 2 * FO * FI + bcol * FI;
    const float* __restrict__ xa  = x + (size_t)(m0 + r16) * FI;

    v8f acc0[MT], acc2[MT], acc3[MT];
#pragma unroll
    for (int t = 0; t < MT; ++t) { acc0[t] = (v8f){}; acc2[t] = (v8f){}; acc3[t] = (v8f){}; }

#pragma unroll 2
    for (int k0 = 0; k0 < FI; k0 += 4) {
        const int ka = k0 + half * 2;
        // ---- load phase: 8 independent b64 loads (immediate offsets) ----
        const v2f b0 = *(const v2f*)(Wt0 + ka);
        const v2f b2 = *(const v2f*)(Wt2 + ka);
        const v2f b3 = *(const v2f*)(Wt3 + ka);
        v2f a[MT];
#pragma unroll
        for (int t = 0; t < MT; ++t)
            a[t] = *(const v2f*)(xa + t * 16 * FI + ka);
        // ---- matrix phase: 15 WMMAs ----
#pragma unroll
        for (int t = 0; t < MT; ++t) {
            acc0[t] = __builtin_amdgcn_wmma_f32_16x16x4_f32(false, a[t], false, b0, (short)0, acc0[t], false, false);
            acc2[t] = __builtin_amdgcn_wmma_f32_16x16x4_f32(false, a[t], false, b2, (short)0, acc2[t], false, false);
            acc3[t] = __builtin_amdgcn_wmma_f32_16x16x4_f32(false, a[t], false, b3, (short)0, acc3[t], false, false);
        }
    }

    // Per-column combined biases (b + chebb): cheb_conv(H=0) contributes only its bias.
    const float c0 = b[0 * FO + bcol] + chebb[0 * FO + bcol];
    const float c2 = b[2 * FO + bcol] + chebb[2 * FO + bcol];
    const float c3 = b[3 * FO + bcol] + chebb[3 * FO + bcol];

#pragma unroll
    for (int t = 0; t < MT; ++t) {
#pragma unroll
        for (int r = 0; r < 8; ++r) {
            const int row = m0 + 16 * t + r + half * 8;
            const float gi = sigm(acc0[t][r] + c0);
            const float gt = tanhf(acc2[t][r] + c2);
            const float go = sigm(acc3[t][r] + c3);
            h1[(size_t)row * FO + bcol] = go * tanhf(gi * gt);
        }
    }
}

// ---------------------------------------------------------------------------
// GCN feature transform: y = h1 @ convW (square F x F), MT=5 strip per wave.
// Also initializes the aggregation buffer with the self-loop term:
//   accum = dinv^2 * y + convb
// ---------------------------------------------------------------------------
template <int F>
__global__ __launch_bounds__(THREADS)
void conv_wmma_kernel(const float* __restrict__ h1,    // n x F
                      const float* __restrict__ Wct,   // F x F (transposed)
                      const float* __restrict__ bc,    // F
                      const float* __restrict__ dinv,  // n
                      float* __restrict__ y,           // n x F
                      float* __restrict__ accum,       // n x F
                      int n)
{
    const int lane   = threadIdx.x & 31;
    const int wave   = threadIdx.x >> 5;
    constexpr int NT = F >> 4;
    const int gw     = blockIdx.x * WAVES_PER_BLOCK + wave;
    const int mtg    = gw / NT;
    const int nt     = gw - mtg * NT;
    const int m0     = mtg * (16 * MT);
    if (m0 >= n) return;

    const int half = lane >> 4;
    const int r16  = lane & 15;
    const int bcol = nt * 16 + r16;

    const float* __restrict__ Wb = Wct + bcol * F;
    const float* __restrict__ ha = h1 + (size_t)(m0 + r16) * F;

    v8f acc[MT];
#pragma unroll
    for (int t = 0; t < MT; ++t) acc[t] = (v8f){};

#pragma unroll 2
    for (int k0 = 0; k0 < F; k0 += 4) {
        const int ka = k0 + half * 2;
        const v2f bm = *(const v2f*)(Wb + ka);
        v2f a[MT];
#pragma unroll
        for (int t = 0; t < MT; ++t)
            a[t] = *(const v2f*)(ha + t * 16 * F + ka);
#pragma unroll
        for (int t = 0; t < MT; ++t)
            acc[t] = __builtin_amdgcn_wmma_f32_16x16x4_f32(false, a[t], false, bm, (short)0, acc[t], false, false);
    }

    const float bias = bc[bcol];
#pragma unroll
    for (int t = 0; t < MT; ++t) {
#pragma unroll
        for (int r = 0; r < 8; ++r) {
            const int row = m0 + 16 * t + r + half * 8;
            const float v  = acc[t][r];
            const float di = dinv[row];
            y[(size_t)row * F + bcol]     = v;
            accum[(size_t)row * F + bcol] = di * di * v + bias;
        }
    }
}

// ---------------------------------------------------------------------------
// Graph helpers
// ---------------------------------------------------------------------------
__global__ void zero_kernel(float* __restrict__ p, int n) {
    int i = blockIdx.x * blockDim.x + threadIdx.x;
    if (i < n) p[i] = 0.0f;
}

__global__ void deg_kernel(const int* __restrict__ dst, float* __restrict__ deg, int e) {
    int i = blockIdx.x * blockDim.x + threadIdx.x;
    if (i < e) atomicAdd(&deg[dst[i]], 1.0f);
}

__global__ void dinv_kernel(const float* __restrict__ deg, float* __restrict__ dinv, int n) {
    int i = blockIdx.x * blockDim.x + threadIdx.x;
    if (i < n) dinv[i] = rsqrtf(deg[i] + 1.0f);     // +1 self loop
}

// accum[dst] += dinv[src]*dinv[dst] * y[src]; one thread per (edge, float4)
__global__ void scatter_kernel(const int* __restrict__ src, const int* __restrict__ dst,
                               const float* __restrict__ dinv, const float* __restrict__ y,
                               float* __restrict__ accum, int e, int fo4)
{
    int gid = blockIdx.x * blockDim.x + threadIdx.x;
    int ed  = gid / fo4;
    if (ed >= e) return;
    int fb = (gid - ed * fo4) * 4;
    int fo = fo4 * 4;
    int s = src[ed], d = dst[ed];
    float c = dinv[s] * dinv[d];
    const float4 v = *(const float4*)(y + (size_t)s * fo + fb);
    float* ad = accum + (size_t)d * fo + fb;
    atomicAdd(ad + 0, c * v.x);
    atomicAdd(ad + 1, c * v.y);
    atomicAdd(ad + 2, c * v.z);
    atomicAdd(ad + 3, c * v.w);
}

// ReLU + BatchNorm1d (eval mode)
__global__ void bn_relu_kernel(const float* __restrict__ acc,
                               const float* __restrict__ gamma, const float* __restrict__ beta,
                               const float* __restrict__ rmean, const float* __restrict__ rvar,
                               float* __restrict__ outp, int total, int fo)
{
    int i = blockIdx.x * blockDim.x + threadIdx.x;
    if (i >= total) return;
    int c = i & (fo - 1);                           // fo is a power of two
    float v = acc[i];
    v = v > 0.0f ? v : 0.0f;
    outp[i] = (v - rmean[c]) * rsqrtf(rvar[c] + 1e-5f) * gamma[c] + beta[c];
}

// Final 16 -> 1 linear
__global__ void final_linear_kernel(const float* __restrict__ h, const float* __restrict__ linW,
                                    const float* __restrict__ linb, float* __restrict__ out, int n)
{
    int i = blockIdx.x * blockDim.x + threadIdx.x;
    if (i >= n) return;
    const float4* hv = (const float4*)(h + (size_t)i * 16);
    const float4* wv = (const float4*)linW;
    float s = 0.0f;
#pragma unroll
    for (int q = 0; q < 4; ++q) {
        float4 a = hv[q], w = wv[q];
        s += a.x * w.x + a.y * w.y + a.z * w.z + a.w * w.w;
    }
    out[i] = s + linb[0];
}

// ---------------------------------------------------------------------------
// Host orchestration
// ---------------------------------------------------------------------------
static inline int gemm_blocks(int N, int fo) {
    int waves = (N / (16 * MT)) * (fo / 16);
    return (waves + WAVES_PER_BLOCK - 1) / WAVES_PER_BLOCK;
}

extern "C" void kernel_launch(void* const* d_in, const int* in_sizes, int n_in,
                              void* d_out, int out_size, void* d_ws, size_t ws_size,
                              hipStream_t stream)
{
    const int F_IN = 32;
    const int N = in_sizes[0] / F_IN;               // 50000
    const int E = in_sizes[2];                      // 800000 (edge_weight length)

    const float* x   = (const float*)d_in[0];
    const int*   ei  = (const int*)d_in[1];
    const int*   src = ei;
    const int*   dst = ei + E;
    // d_in[2] (edge_weight) is mathematically unused: cheb_conv runs on H=0.

    const int fis[4] = {32, 128, 64, 32};
    const int fos[4] = {128, 64, 32, 16};

    // Workspace: 3 rotating N x 128 buffers + deg + dinv + transposed weights
    const size_t big = (size_t)N * 128;
    float* B[3];
    B[0] = (float*)d_ws;
    B[1] = B[0] + big;
    B[2] = B[1] + big;
    float* deg  = B[2] + big;
    float* dinv = deg + N;
    float* Wt   = dinv + N;                         // 3 * 128 * 128 max
    float* Wct  = Wt + 3 * 128 * 128;               // 128 * 128 max

    // degrees / dinv (graph is identical for every layer)
    zero_kernel<<<(N + THREADS - 1) / THREADS, THREADS, 0, stream>>>(deg, N);
    deg_kernel<<<(E + THREADS - 1) / THREADS, THREADS, 0, stream>>>(dst, deg, E);
    dinv_kernel<<<(N + THREADS - 1) / THREADS, THREADS, 0, stream>>>(deg, dinv, N);

    const float* hin = x;
    int ih = 0;                                     // rotating buffer role index
    for (int l = 0; l < 4; ++l) {
        const int fi = fis[l], fo = fos[l];
        const int base = 3 + 10 * l;
        const float* W     = (const float*)d_in[base + 0];
        const float* bg    = (const float*)d_in[base + 1];
        const float* chebb = (const float*)d_in[base + 3];
        const float* Wc    = (const float*)d_in[base + 4];
        const float* bc    = (const float*)d_in[base + 5];
        const float* gamma = (const float*)d_in[base + 6];
        const float* beta  = (const float*)d_in[base + 7];
        const float* rmean = (const float*)d_in[base + 8];
        const float* rvar  = (const float*)d_in[base + 9];

        float* h1 = B[ih];
        float* yb = B[(ih + 1) % 3];
        float* ac = B[(ih + 2) % 3];                // == previous hin buffer, dead after gates

        // 0) transpose weights for contiguous B-fragment loads
        {
            int tg = 3 * fi * fo;
            transpose_gates_kernel<<<(tg + THREADS - 1) / THREADS, THREADS, 0, stream>>>(W, Wt, fi, fo);
            int tc = fo * fo;
            transpose_kernel<<<(tc + THREADS - 1) / THREADS, THREADS, 0, stream>>>(Wc, Wct, fo, fo);
        }
        // 1) gate GEMMs + LSTM activations (WMMA, 80x16 strip per wave)
        {
            int blocks = gemm_blocks(N, fo);
            switch (l) {
            case 0: gates_wmma_kernel< 32,128><<<blocks, THREADS, 0, stream>>>(hin, Wt, bg, chebb, h1, N); break;
            case 1: gates_wmma_kernel<128, 64><<<blocks, THREADS, 0, stream>>>(hin, Wt, bg, chebb, h1, N); break;
            case 2: gates_wmma_kernel< 64, 32><<<blocks, THREADS, 0, stream>>>(hin, Wt, bg, chebb, h1, N); break;
            default:gates_wmma_kernel< 32, 16><<<blocks, THREADS, 0, stream>>>(hin, Wt, bg, chebb, h1, N); break;
            }
        }
        // 2) GCN transform + self-loop init (WMMA)
        {
            int blocks = gemm_blocks(N, fo);
            switch (l) {
            case 0: conv_wmma_kernel<128><<<blocks, THREADS, 0, stream>>>(h1, Wct, bc, dinv, yb, ac, N); break;
            case 1: conv_wmma_kernel< 64><<<blocks, THREADS, 0, stream>>>(h1, Wct, bc, dinv, yb, ac, N); break;
            case 2: conv_wmma_kernel< 32><<<blocks, THREADS, 0, stream>>>(h1, Wct, bc, dinv, yb, ac, N); break;
            default:conv_wmma_kernel< 16><<<blocks, THREADS, 0, stream>>>(h1, Wct, bc, dinv, yb, ac, N); break;
            }
        }
        // 3) edge scatter (float atomics, L2-resident gathers)
        {
            int fo4   = fo / 4;
            int total = E * fo4;
            scatter_kernel<<<(total + THREADS - 1) / THREADS, THREADS, 0, stream>>>(
                src, dst, dinv, yb, ac, E, fo4);
        }
        // 4) ReLU + BatchNorm (eval); write into h1 buffer (dead) = next layer input
        {
            int total = N * fo;
            bn_relu_kernel<<<(total + THREADS - 1) / THREADS, THREADS, 0, stream>>>(
                ac, gamma, beta, rmean, rvar, h1, total, fo);
        }
        hin = h1;
        ih = (ih + 1) % 3;
    }

    const float* linW = (const float*)d_in[43];
    const float* linb = (const float*)d_in[44];
    final_linear_kernel<<<(N + THREADS - 1) / THREADS, THREADS, 0, stream>>>(
        hin, linW, linb, (float*)d_out, N);
}